// CPDP_AM_net_SGBS_7009386627334
// MI455X (gfx1250) — compile-verified
//
#include <hip/hip_runtime.h>
#include <hip/hip_bf16.h>

// Problem constants (fixed by the reference's setup_inputs).
#define BATCH 256
#define NKEY  512
#define DMODEL 512
#define LDIM  3
#define NHEAD 16

typedef float v2f __attribute__((ext_vector_type(2)));
typedef float v8f __attribute__((ext_vector_type(8)));

// ---------------------------------------------------------------------------
// Projection GEMM: Y[m,n] = sum_k X[m,k] * W[n,k] + bias[n]
// M=BATCH(256), N=DMODEL(512), K=DMODEL(512), full fp32 via V_WMMA_F32_16X16X4_F32.
// One wave per 16x16 output tile.
//  A layout (16x4 f32): lane l holds A[l&15][2*half], A[l&15][2*half+1]  (half = l>>4)
//  B layout (4x16 f32): lane l holds B[2*half][l&15], B[2*half+1][l&15]
//  -> both are contiguous float2 loads at k0 + 2*half.
//  C/D layout: VGPR r, lane l -> (M = r + 8*half, N = l&15).
// ---------------------------------------------------------------------------
__global__ __launch_bounds__(32)
void proj_wmma_kernel(const float* __restrict__ X,   // (256, 512)
                      const float* __restrict__ W,   // (512, 512), used as W^T
                      const float* __restrict__ bias,// (512)
                      float* __restrict__ Y)         // (256, 512)
{
    const int lane = threadIdx.x & 31;
    const int half = lane >> 4;
    const int lrow = lane & 15;
    const int m0 = blockIdx.y * 16;
    const int n0 = blockIdx.x * 16;

    const float* arow = X + (size_t)(m0 + lrow) * DMODEL + 2 * half;
    const float* brow = W + (size_t)(n0 + lrow) * DMODEL + 2 * half;

    v8f acc = {};
#pragma unroll 8
    for (int k = 0; k < DMODEL; k += 4) {
        v2f a = *(const v2f*)(arow + k);
        v2f b = *(const v2f*)(brow + k);
        acc = __builtin_amdgcn_wmma_f32_16x16x4_f32(
            /*neg_a=*/false, a, /*neg_b=*/false, b,
            /*c_mod=*/(short)0, acc, /*reuse_a=*/false, /*reuse_b=*/false);
    }

    const float bn = bias[n0 + lrow];
#pragma unroll
    for (int r = 0; r < 8; ++r) {
        Y[(size_t)(m0 + r + 8 * half) * DMODEL + n0 + lrow] = acc[r] + bn;
    }
}

// ---------------------------------------------------------------------------
// 16-head attention layer (layers 0 and 1). One block per batch, 256 threads.
// q_out[b, :] = softmax_n( (q . K_l[b,n]) / sqrt(32), masked ) @ V_l[b]
// ---------------------------------------------------------------------------
__global__ __launch_bounds__(256)
void attn16_kernel(const float* __restrict__ q_in,      // (B, D)
                   const float* __restrict__ K_att,     // (B, N, 3D)
                   const float* __restrict__ V_att,     // (B, N, 3D)
                   const unsigned char* __restrict__ mask, // (B, N) bool
                   float* __restrict__ q_out,           // (B, D)
                   int layer)
{
    __shared__ float q_s[DMODEL];
    __shared__ float sc[NHEAD * NKEY];   // scores -> unnormalized weights
    __shared__ float red[256];
    __shared__ float hmax[NHEAD];
    __shared__ float hinv[NHEAD];

    const int b    = blockIdx.x;
    const int t    = threadIdx.x;
    const int lane = t & 31;
    const int wave = t >> 5;

    // stage q into LDS
    q_s[t]       = q_in[(size_t)b * DMODEL + t];
    q_s[t + 256] = q_in[(size_t)b * DMODEL + t + 256];
    __syncthreads();

    const size_t row_stride = (size_t)LDIM * DMODEL;                 // 1536 floats
    const float* Kbase = K_att + (size_t)b * NKEY * row_stride + (size_t)layer * DMODEL;
    const unsigned char* mrow = mask + (size_t)b * NKEY;
    const float scale = 0.17677669529663687f;                        // 1/sqrt(32)

    // ---- scores: one wave per key; lanes cover D, per-head shfl reduction ----
    for (int n = wave; n < NKEY; n += 8) {
        const float* krow = Kbase + (size_t)n * row_stride;
        float p[4];
#pragma unroll
        for (int j = 0; j < 4; ++j) {
            const int d0 = lane * 4 + 128 * j;                       // stays inside head d0>>5
            const float4 kv = *(const float4*)(krow + d0);
            const float4 qv = *(const float4*)(q_s + d0);
            p[j] = kv.x * qv.x + kv.y * qv.y + kv.z * qv.z + kv.w * qv.w;
            // reduce over the 8 lanes sharing head (lane>>3)
            p[j] += __shfl_xor(p[j], 1);
            p[j] += __shfl_xor(p[j], 2);
            p[j] += __shfl_xor(p[j], 4);
        }
        if ((lane & 7) == 0) {
            const bool m = (mrow[n] != 0);
#pragma unroll
            for (int j = 0; j < 4; ++j) {
                const int h = (lane >> 3) + 4 * j;
                sc[h * NKEY + n] = m ? -1.0e9f : p[j] * scale;
            }
        }
    }
    __syncthreads();

    // ---- stable softmax per head (16 threads per head) ----
    const int h   = t >> 4;
    const int sub = t & 15;
    float m = -3.0e38f;
    for (int i = 0; i < 32; ++i)
        m = fmaxf(m, sc[h * NKEY + sub + 16 * i]);
    red[t] = m;
    __syncthreads();
    if (t < NHEAD) {
        float mm = red[t * 16];
        for (int i = 1; i < 16; ++i) mm = fmaxf(mm, red[t * 16 + i]);
        hmax[t] = mm;
    }
    __syncthreads();

    const float hm = hmax[h];
    float s = 0.0f;
    for (int i = 0; i < 32; ++i) {
        const int idx = h * NKEY + sub + 16 * i;
        const float w = expf(sc[idx] - hm);                          // masked -> exp(<-1e8) == 0
        sc[idx] = w;                                                 // keep unnormalized
        s += w;
    }
    red[t] = s;
    __syncthreads();
    if (t < NHEAD) {
        float ss = 0.0f;
        for (int i = 0; i < 16; ++i) ss += red[t * 16 + i];
        hinv[t] = 1.0f / ss;                                         // n=0 never masked -> ss >= 1
    }
    __syncthreads();

    // ---- weighted sum over V: thread t owns output pair d0 = 2t ----
    const int d0 = t * 2;
    const int oh = d0 >> 5;
    const float* Vcol = V_att + (size_t)b * NKEY * row_stride + (size_t)layer * DMODEL + d0;
    const float* wrow = &sc[oh * NKEY];
    float ax = 0.0f, ay = 0.0f;
    for (int n = 0; n < NKEY; ++n) {
        const float w  = wrow[n];
        const float2 v = *(const float2*)(Vcol + (size_t)n * row_stride);
        ax += w * v.x;
        ay += w * v.y;
    }
    const float inv = hinv[oh];
    float2 o; o.x = ax * inv; o.y = ay * inv;
    *(float2*)(q_out + (size_t)b * DMODEL + d0) = o;
}

// ---------------------------------------------------------------------------
// Final layer: single head over full D, score = 10*tanh(q.K / sqrt(512)),
// mask -> -1e9, softmax over n. Output is the weights (B, N). V unused.
// ---------------------------------------------------------------------------
__global__ __launch_bounds__(256)
void attn_final_kernel(const float* __restrict__ q_in,   // (B, D)
                       const float* __restrict__ K_att,  // (B, N, 3D)
                       const unsigned char* __restrict__ mask,
                       float* __restrict__ out)          // (B, N)
{
    __shared__ float q_s[DMODEL];
    __shared__ float sc[NKEY];
    __shared__ float red[256];

    const int b    = blockIdx.x;
    const int t    = threadIdx.x;
    const int lane = t & 31;
    const int wave = t >> 5;

    q_s[t]       = q_in[(size_t)b * DMODEL + t];
    q_s[t + 256] = q_in[(size_t)b * DMODEL + t + 256];
    __syncthreads();

    const size_t row_stride = (size_t)LDIM * DMODEL;
    const float* Kbase = K_att + (size_t)b * NKEY * row_stride + 2 * DMODEL; // layer 2
    const unsigned char* mrow = mask + (size_t)b * NKEY;
    const float scale = 0.044194173824159216f;                       // 1/sqrt(512)

    // one wave per key, full 32-lane reduction
    for (int n = wave; n < NKEY; n += 8) {
        const float* krow = Kbase + (size_t)n * row_stride;
        float p = 0.0f;
#pragma unroll
        for (int j = 0; j < 4; ++j) {
            const int d0 = lane * 4 + 128 * j;
            const float4 kv = *(const float4*)(krow + d0);
            const float4 qv = *(const float4*)(q_s + d0);
            p += kv.x * qv.x + kv.y * qv.y + kv.z * qv.z + kv.w * qv.w;
        }
        p += __shfl_xor(p, 1);
        p += __shfl_xor(p, 2);
        p += __shfl_xor(p, 4);
        p += __shfl_xor(p, 8);
        p += __shfl_xor(p, 16);
        if (lane == 0) {
            const float sv = 10.0f * tanhf(p * scale);
            sc[n] = (mrow[n] != 0) ? -1.0e9f : sv;
        }
    }
    __syncthreads();

    // block-wide stable softmax over 512 scores
    red[t] = fmaxf(sc[t], sc[t + 256]);
    __syncthreads();
    for (int off = 128; off > 0; off >>= 1) {
        if (t < off) red[t] = fmaxf(red[t], red[t + off]);
        __syncthreads();
    }
    const float mx = red[0];
    __syncthreads();

    const float e0 = expf(sc[t] - mx);
    const float e1 = expf(sc[t + 256] - mx);
    red[t] = e0 + e1;
    __syncthreads();
    for (int off = 128; off > 0; off >>= 1) {
        if (t < off) red[t] += red[t + off];
        __syncthreads();
    }
    const float inv = 1.0f / red[0];

    out[(size_t)b * NKEY + t]       = e0 * inv;
    out[(size_t)b * NKEY + t + 256] = e1 * inv;
}

// ---------------------------------------------------------------------------
// Driver: 3 attention stages with the 3 shared-weight projections between.
// ---------------------------------------------------------------------------
extern "C" void kernel_launch(void* const* d_in, const int* in_sizes, int n_in,
                              void* d_out, int out_size, void* d_ws, size_t ws_size,
                              hipStream_t stream) {
    const float*         query = (const float*)d_in[0];          // (B,1,D)
    const float*         K_att = (const float*)d_in[1];          // (B,N,3D)
    const float*         V_att = (const float*)d_in[2];          // (B,N,3D)
    const unsigned char* mask  = (const unsigned char*)d_in[3];  // (B,N) bool (1 byte)
    const float*         W0_w  = (const float*)d_in[4];          // (D,D)
    const float*         W0_b  = (const float*)d_in[5];          // (D)
    const float*         Wq_w  = (const float*)d_in[6];          // (D,D)
    const float*         Wq_b  = (const float*)d_in[7];          // (D)
    float*               out   = (float*)d_out;                  // (B,N)

    float* wsA = (float*)d_ws;                    // (B, D)
    float* wsB = wsA + (size_t)BATCH * DMODEL;    // (B, D)

    const dim3 attnGrid(BATCH);
    const dim3 attnBlock(256);
    const dim3 projGrid(DMODEL / 16, BATCH / 16); // (32, 16) tiles
    const dim3 projBlock(32);

    // layer 0
    attn16_kernel<<<attnGrid, attnBlock, 0, stream>>>(query, K_att, V_att, mask, wsA, 0);
    proj_wmma_kernel<<<projGrid, projBlock, 0, stream>>>(wsA, W0_w, W0_b, wsB);
    // layer 1
    attn16_kernel<<<attnGrid, attnBlock, 0, stream>>>(wsB, K_att, V_att, mask, wsA, 1);
    proj_wmma_kernel<<<projGrid, projBlock, 0, stream>>>(wsA, W0_w, W0_b, wsB);
    // final query projection
    proj_wmma_kernel<<<projGrid, projBlock, 0, stream>>>(wsB, Wq_w, Wq_b, wsA);
    // layer 2: clipped single-head attention -> weights
    attn_final_kernel<<<attnGrid, attnBlock, 0, stream>>>(wsA, K_att, mask, out);
}